// GNNModel_1202590843251
// MI455X (gfx1250) — compile-verified
//
#include <hip/hip_runtime.h>
#include <cmath>

typedef __attribute__((ext_vector_type(2))) float v2f;
typedef __attribute__((ext_vector_type(8))) float v8f;

// ---------- monotonic float<->uint key for integer atomicMax on floats ----------
static __device__ __forceinline__ unsigned fkey(float x) {
    unsigned u = __float_as_uint(x);
    return (u & 0x80000000u) ? ~u : (u | 0x80000000u);
}
static __device__ __forceinline__ float funkey(unsigned k) {
    unsigned u = (k & 0x80000000u) ? (k ^ 0x80000000u) : ~k;
    return __uint_as_float(u);
}

// ---------- generic fill ----------
__global__ void fill_f32(float* __restrict__ p, float v, long long n) {
    long long i = (long long)blockIdx.x * blockDim.x + threadIdx.x;
    if (i < n) p[i] = v;
}

// ---------- fp32 WMMA GEMM: C[MxN] = A[MxK] * B[KxN], M%16==0, N%16==0, K%4==0 ----------
// One wave per 16x16 output tile; 16x16x4 f32 WMMA per K-step.
__global__ void gemm_wmma_f32(const float* __restrict__ A, const float* __restrict__ B,
                              float* __restrict__ C, int M, int K, int N) {
    const int wavesPerBlk = blockDim.x >> 5;
    const int wave = blockIdx.x * wavesPerBlk + (threadIdx.x >> 5);
    const int lane = threadIdx.x & 31;
    const int tilesN = N >> 4;
    const int tm = wave / tilesN;
    const int tn = wave - tm * tilesN;
    if (tm * 16 >= M) return;
    const int  l16 = lane & 15;
    const bool hi  = lane >= 16;
    const size_t row = (size_t)(tm * 16 + l16);   // A-row for this lane
    const int    col = tn * 16 + l16;             // B/C column for this lane
    v8f acc = {};
    for (int k = 0; k < K; k += 4) {
        const int ka = k + (hi ? 2 : 0);
        // A 16x4 layout: v0=K(ka), v1=K(ka+1) -> single 8B vector load per lane
        v2f a = *(const v2f*)(A + row * (size_t)K + ka);
        v2f b;
        b.x = B[(size_t)ka * N + col];
        b.y = B[(size_t)(ka + 1) * N + col];
        acc = __builtin_amdgcn_wmma_f32_16x16x4_f32(false, a, false, b,
                                                    (short)0, acc, false, false);
    }
    const size_t rbase = (size_t)tm * 16 + (hi ? 8 : 0);
#pragma unroll
    for (int v = 0; v < 8; ++v)
        C[(rbase + v) * (size_t)N + col] = acc[v];
}

// ---------- scalar GEMM for small/odd K or N (layers 1 and 3) ----------
__global__ void gemm_scalar(const float* __restrict__ A, const float* __restrict__ B,
                            float* __restrict__ C, long long M, int K, int N) {
    long long t = (long long)blockIdx.x * blockDim.x + threadIdx.x;
    if (t >= M * (long long)N) return;
    long long m = t / N;
    int n = (int)(t - m * N);
    float acc = 0.f;
    for (int k = 0; k < K; ++k)
        acc = fmaf(A[m * K + k], B[(size_t)k * N + n], acc);
    C[t] = acc;
}

// ---------- per-node attention coefficients: as[i]=h_i.a_src, ad[i]=h_i.a_dst ----------
__global__ void node_alpha(const float* __restrict__ h, const float* __restrict__ a_src,
                           const float* __restrict__ a_dst, float* __restrict__ as_,
                           float* __restrict__ ad_, long long N, int F) {
    long long node = ((long long)blockIdx.x * blockDim.x + threadIdx.x) >> 5;
    int lane = threadIdx.x & 31;
    if (node >= N) return;
    float s = 0.f, d = 0.f;
    for (int f = lane; f < F; f += 32) {
        float hv = h[node * F + f];
        s = fmaf(hv, a_src[f], s);
        d = fmaf(hv, a_dst[f], d);
    }
    for (int off = 16; off; off >>= 1) {
        s += __shfl_down(s, off, 32);
        d += __shfl_down(d, off, 32);
    }
    if (lane == 0) { as_[node] = s; ad_[node] = d; }
}

// ---------- edge pass 1: leaky-relu logits + per-target running max ----------
__global__ void edge_logits(const long long* __restrict__ src, const long long* __restrict__ dst,
                            const float* __restrict__ as_, const float* __restrict__ ad_,
                            float* __restrict__ ebuf, unsigned* __restrict__ mkey,
                            long long E, long long N) {
    long long i = (long long)blockIdx.x * blockDim.x + threadIdx.x;
    if (i >= E + N) return;
    long long s, d;
    if (i < E) { s = src[i]; d = dst[i]; } else { s = d = i - E; }
    float e = as_[s] + ad_[d];
    e = e > 0.f ? e : 0.2f * e;
    ebuf[i] = e;
    atomicMax(&mkey[d], fkey(e));
}

// ---------- edge pass 2: ex = exp(e - m[dst]); denom[dst] += ex ----------
__global__ void edge_expsum(const long long* __restrict__ dst, float* __restrict__ ebuf,
                            const unsigned* __restrict__ mkey, float* __restrict__ denom,
                            long long E, long long N) {
    long long i = (long long)blockIdx.x * blockDim.x + threadIdx.x;
    if (i >= E + N) return;
    long long d = (i < E) ? dst[i] : (i - E);
    float ex = expf(ebuf[i] - funkey(mkey[d]));
    ebuf[i] = ex;
    unsafeAtomicAdd(&denom[d], ex);
}

// ---------- edge pass 3 (F==128): wave per edge, float4 gather + 4 fp32 atomics/lane ----------
__global__ void edge_scatter128(const long long* __restrict__ src, const long long* __restrict__ dst,
                                const float* __restrict__ ebuf, const float* __restrict__ denom,
                                const float* __restrict__ h, float* __restrict__ out,
                                long long E, long long N) {
    long long w = (long long)blockIdx.x * (blockDim.x >> 5) + (threadIdx.x >> 5);
    int lane = threadIdx.x & 31;
    if (w >= E + N) return;
    long long s, d;
    if (w < E) { s = src[w]; d = dst[w]; } else { s = d = w - E; }
    float alpha = ebuf[w] / (denom[d] + 1e-16f);
    float4 hv = *((const float4*)(h + s * 128) + lane);
    float* o = out + d * 128 + lane * 4;
    unsafeAtomicAdd(o + 0, alpha * hv.x);
    unsafeAtomicAdd(o + 1, alpha * hv.y);
    unsafeAtomicAdd(o + 2, alpha * hv.z);
    unsafeAtomicAdd(o + 3, alpha * hv.w);
}

// ---------- edge pass 3 (small F): thread per edge ----------
__global__ void edge_scatter_small(const long long* __restrict__ src, const long long* __restrict__ dst,
                                   const float* __restrict__ ebuf, const float* __restrict__ denom,
                                   const float* __restrict__ h, float* __restrict__ out,
                                   long long E, long long N, int F) {
    long long i = (long long)blockIdx.x * blockDim.x + threadIdx.x;
    if (i >= E + N) return;
    long long s, d;
    if (i < E) { s = src[i]; d = dst[i]; } else { s = d = i - E; }
    float alpha = ebuf[i] / (denom[d] + 1e-16f);
    for (int f = 0; f < F; ++f)
        unsafeAtomicAdd(&out[d * F + f], alpha * h[s * F + f]);
}

// ---------- epilogue: x = relu(x + b) ----------
__global__ void bias_relu(float* __restrict__ p, const float* __restrict__ b,
                          long long n, int F) {
    long long i = (long long)blockIdx.x * blockDim.x + threadIdx.x;
    if (i >= n) return;
    float v = p[i] + b[(int)(i % F)];
    p[i] = fmaxf(v, 0.f);
}

// ---------- final layer: out = log_softmax(acc + b), F<=8 ----------
__global__ void bias_logsoftmax(const float* __restrict__ acc, const float* __restrict__ b,
                                float* __restrict__ out, long long N, int F) {
    long long n = (long long)blockIdx.x * blockDim.x + threadIdx.x;
    if (n >= N) return;
    float v[8];
    float mx = -INFINITY;
    for (int f = 0; f < F; ++f) { v[f] = acc[n * F + f] + b[f]; mx = fmaxf(mx, v[f]); }
    float ssum = 0.f;
    for (int f = 0; f < F; ++f) ssum += expf(v[f] - mx);
    float lse = mx + logf(ssum);
    for (int f = 0; f < F; ++f) out[n * F + f] = v[f] - lse;
}

static inline unsigned blks(long long n, int t) { return (unsigned)((n + t - 1) / t); }

extern "C" void kernel_launch(void* const* d_in, const int* in_sizes, int n_in,
                              void* d_out, int out_size, void* d_ws, size_t ws_size,
                              hipStream_t stream) {
    const float*     x    = (const float*)d_in[0];
    const long long* eidx = (const long long*)d_in[1];   // int64 [2, E]
    const float* W1  = (const float*)d_in[2];
    const float* as1 = (const float*)d_in[3];
    const float* ad1 = (const float*)d_in[4];
    const float* b1  = (const float*)d_in[5];
    const float* W2  = (const float*)d_in[6];
    const float* as2 = (const float*)d_in[7];
    const float* ad2 = (const float*)d_in[8];
    const float* b2  = (const float*)d_in[9];
    const float* W3  = (const float*)d_in[10];
    const float* as3 = (const float*)d_in[11];
    const float* ad3 = (const float*)d_in[12];
    const float* b3  = (const float*)d_in[13];

    const int IN = 13, HID = 128, OUT = 5;
    const long long N  = in_sizes[0] / IN;
    const long long E  = in_sizes[1] / 2;
    const long long ET = E + N;                 // edges + self loops
    const long long* srcp = eidx;
    const long long* dstp = eidx + E;

    // workspace layout (floats)
    float* ws = (float*)d_ws;
    size_t off = 0;
    float*    hw    = ws + off; off += (size_t)N * HID;   // GEMM output h
    float*    accb  = ws + off; off += (size_t)N * HID;   // attention accumulator / next input
    float*    h3acc = ws + off; off += (size_t)N * OUT;   // layer-3 accumulator
    float*    asb   = ws + off; off += (size_t)N;
    float*    adb   = ws + off; off += (size_t)N;
    float*    denom = ws + off; off += (size_t)N;
    unsigned* mkey  = (unsigned*)(ws + off); off += (size_t)N;
    float*    ebuf  = ws + off; off += (size_t)ET;

    const int T = 256;

    // ---- helper macro: run attention (softmax aggregation) on h -> outbuf ----
    #define RUN_ATTN(H, OUTBUF, ASRC, ADST, F)                                                  \
        do {                                                                                    \
            fill_f32<<<blks((long long)N * (F), T), T, 0, stream>>>(OUTBUF, 0.f, (long long)N * (F)); \
            fill_f32<<<blks(N, T), T, 0, stream>>>(denom, 0.f, N);                              \
            fill_f32<<<blks(N, T), T, 0, stream>>>((float*)mkey, 0.f, N); /* key 0 == -inf */   \
            node_alpha<<<blks(N * 32, T), T, 0, stream>>>(H, ASRC, ADST, asb, adb, N, F);       \
            edge_logits<<<blks(ET, T), T, 0, stream>>>(srcp, dstp, asb, adb, ebuf, mkey, E, N); \
            edge_expsum<<<blks(ET, T), T, 0, stream>>>(dstp, ebuf, mkey, denom, E, N);          \
            if ((F) == 128)                                                                     \
                edge_scatter128<<<blks(ET * 32, T), T, 0, stream>>>(srcp, dstp, ebuf, denom, H, OUTBUF, E, N); \
            else                                                                                \
                edge_scatter_small<<<blks(ET, T), T, 0, stream>>>(srcp, dstp, ebuf, denom, H, OUTBUF, E, N, F); \
        } while (0)

    // ---------------- layer 1: h = x @ W1 (K=13, scalar), attention, relu ----------------
    gemm_scalar<<<blks(N * HID, T), T, 0, stream>>>(x, W1, hw, N, IN, HID);
    RUN_ATTN(hw, accb, as1, ad1, HID);
    bias_relu<<<blks(N * HID, T), T, 0, stream>>>(accb, b1, N * HID, HID);

    // ---------------- layer 2: h = accb @ W2 (128x128, fp32 WMMA), attention, relu -------
    {
        long long waves = (N / 16) * (HID / 16);
        gemm_wmma_f32<<<blks(waves * 32, T), T, 0, stream>>>(accb, W2, hw, (int)N, HID, HID);
    }
    RUN_ATTN(hw, accb, as2, ad2, HID);   // accb reused: GEMM consumed it before this zero+scatter
    bias_relu<<<blks(N * HID, T), T, 0, stream>>>(accb, b2, N * HID, HID);

    // ---------------- layer 3: h = accb @ W3 (K=128, N=5), attention, log_softmax --------
    gemm_scalar<<<blks(N * OUT, T), T, 0, stream>>>(accb, W3, hw, N, HID, OUT);
    RUN_ATTN(hw, h3acc, as3, ad3, OUT);
    bias_logsoftmax<<<blks(N, T), T, 0, stream>>>(h3acc, b3, (float*)d_out, N, OUT);

    #undef RUN_ATTN
}